// PermutedCycleAttention_53635551592657
// MI455X (gfx1250) — compile-verified
//
#include <hip/hip_runtime.h>

typedef float v2f __attribute__((ext_vector_type(2)));
typedef float v8f __attribute__((ext_vector_type(8)));

#define Tdim 2048
#define Cdim 512
#define Hn   8
#define DHd  64
#define WINr 48
#define WSZ  (2 * WINr + 1) /* 97 */
#define NEG_BIG (-3.0e38f)

// attention tiling: 16 permuted rows per block, 112-row K/V slab in LDS
#define NT     16
#define NROWS  (NT + 2 * WINr) /* 112 */
#define ABLK   (NT * 32)       /* 512 threads = 16 wave32 */

// ---------------------------------------------------------------------------
// FP32 WMMA GEMM: C[M,N] = A[M,K] * B[K,N], row-major.
// grid = (N/128, M/128), block = 256 threads (8 wave32).
// Wave w computes rows [blockIdx.y*128 + w*16, +16) over a 128-wide N strip
// (8 WMMA 16x16 tiles), K-loop in steps of 4 using V_WMMA_F32_16X16X4_F32.
// ---------------------------------------------------------------------------
__global__ void __launch_bounds__(256)
gemm_f32_wmma(const float* __restrict__ A, const float* __restrict__ B,
              float* __restrict__ C, int K, int N) {
  const int lane = threadIdx.x & 31;
  const int wave = threadIdx.x >> 5;
  const int m0 = blockIdx.y * 128 + wave * 16;
  const int n0 = blockIdx.x * 128;

  const int arow  = m0 + (lane & 15);     // A-fragment row for this lane
  const int khalf = (lane >> 4) << 1;     // 0 (lanes 0-15) or 2 (lanes 16-31)
  const int bcol  = n0 + (lane & 15);     // B-fragment column for this lane

  v8f acc[8] = {};

  for (int k = 0; k < K; k += 4) {
    const float* ap = A + (size_t)arow * K + (k + khalf);
    v2f a;
    a.x = ap[0];
    a.y = ap[1];
    const float* bp = B + (size_t)(k + khalf) * N + bcol;
#pragma unroll
    for (int tn = 0; tn < 8; ++tn) {
      v2f b;
      b.x = bp[tn * 16];
      b.y = bp[tn * 16 + N];
      acc[tn] = __builtin_amdgcn_wmma_f32_16x16x4_f32(
          false, a, false, b, (short)0, acc[tn], false, false);
    }
  }

  const int crow = m0 + ((lane >> 4) << 3);  // +8 for the upper half-wave
  const int ccol = n0 + (lane & 15);
#pragma unroll
  for (int tn = 0; tn < 8; ++tn) {
#pragma unroll
    for (int v = 0; v < 8; ++v) {
      C[(size_t)(crow + v) * N + (ccol + tn * 16)] = acc[tn][v];
    }
  }
}

// ---------------------------------------------------------------------------
// Permuted sliding-window attention with LDS-staged K/V slab.
// Block = 512 threads (16 wave32), handles head h, permuted slots
// [t0, t0+16). The 97-wide windows of these rows all fall inside slab rows
// [t0-48, t0+63], i.e. 112 rows. K/V slab (2 x 112 x 64 f32 = 56 KB) is
// staged with CDNA5 GLOBAL_LOAD_ASYNC_TO_LDS_B64 (ASYNCcnt), then each wave
// computes one row: scores via wave32 butterfly reductions, masked softmax,
// weighted V accumulation -- all K/V reads hit LDS (full-row 256B, 64-bank
// conflict free).
// ---------------------------------------------------------------------------
__global__ void __launch_bounds__(ABLK)
attn_perm_window(const float* __restrict__ qkv, const int* __restrict__ perms,
                 float* __restrict__ y) {
  __shared__ float ks[NROWS][DHd];
  __shared__ float vs[NROWS][DHd];
  __shared__ int   ps[NROWS];

  const int lane = threadIdx.x & 31;
  const int wave = threadIdx.x >> 5;  // 0..15
  const int h  = blockIdx.y;
  const int t0 = blockIdx.x * NT;
  const int* __restrict__ perm = perms + h * Tdim;

  // ---- stage permutation slab (original indices for slab rows) ----
  for (int s = threadIdx.x; s < NROWS; s += ABLK) {
    const int p = t0 - WINr + s;
    const int idx = p < 0 ? 0 : (p > Tdim - 1 ? Tdim - 1 : p);
    ps[s] = perm[idx];
  }
  __syncthreads();

  // ---- async-stage K/V slab: wave w loads rows w, w+16, ... (7 rows) ----
  for (int s = wave; s < NROWS; s += NT) {
    const int pj = ps[s];
    const unsigned long long kg =
        (unsigned long long)(qkv + (size_t)pj * (3 * Cdim) + Cdim + h * DHd +
                             2 * lane);
    const unsigned long long vg =
        (unsigned long long)(qkv + (size_t)pj * (3 * Cdim) + 2 * Cdim +
                             h * DHd + 2 * lane);
    const unsigned kl = (unsigned)(size_t)&ks[s][2 * lane];
    const unsigned vl = (unsigned)(size_t)&vs[s][2 * lane];
    asm volatile("global_load_async_to_lds_b64 %0, %1, off"
                 :
                 : "v"(kl), "v"(kg)
                 : "memory");
    asm volatile("global_load_async_to_lds_b64 %0, %1, off"
                 :
                 : "v"(vl), "v"(vg)
                 : "memory");
  }
  asm volatile("s_wait_asynccnt 0x0" ::: "memory");
  __syncthreads();

  // ---- per-wave row computation ----
  const int t  = t0 + wave;
  const int pt = ps[wave + WINr];  // == perm[t]

  const float2 q =
      *(const float2*)(qkv + (size_t)pt * (3 * Cdim) + h * DHd + 2 * lane);

  float s4[4];
  float e4[4];

  // phase 1: masked scores over the 97-wide window
#pragma unroll
  for (int slot = 0; slot < 4; ++slot) {
    const int jbase = slot << 5;
    const int nj = (WSZ - jbase) < 32 ? (WSZ - jbase) : 32;  // 32,32,32,1
    float ssc = NEG_BIG;
    for (int j2 = 0; j2 < nj; ++j2) {
      const int j  = jbase + j2;
      const int pi = t - WINr + j;
      const int sr = wave + j;  // slab row
      const int pj = ps[sr];
      const bool keep = (pi >= 0) && (pi < Tdim) && (pj <= pt);
      const float2 kv = *(const float2*)&ks[sr][2 * lane];
      float part = q.x * kv.x + q.y * kv.y;
#pragma unroll
      for (int off = 16; off > 0; off >>= 1) part += __shfl_xor(part, off, 32);
      const float sc = keep ? part * 0.125f : NEG_BIG;  // /sqrt(64)
      if (j2 == lane) ssc = sc;  // park score j at lane (j&31), slot j>>5
    }
    s4[slot] = ssc;
  }

  // softmax
  float m = fmaxf(fmaxf(s4[0], s4[1]), fmaxf(s4[2], s4[3]));
#pragma unroll
  for (int off = 16; off > 0; off >>= 1) m = fmaxf(m, __shfl_xor(m, off, 32));
  float sum = 0.0f;
#pragma unroll
  for (int slot = 0; slot < 4; ++slot) {
    float ex = __expf(s4[slot] - m);
    if (s4[slot] <= NEG_BIG) ex = 0.0f;  // exact zero for masked entries
    e4[slot] = ex;
    sum += ex;
  }
#pragma unroll
  for (int off = 16; off > 0; off >>= 1) sum += __shfl_xor(sum, off, 32);
  const float inv = 1.0f / sum;

  // phase 2: weighted V accumulation from LDS
  float2 acc;
  acc.x = 0.0f;
  acc.y = 0.0f;
#pragma unroll
  for (int slot = 0; slot < 4; ++slot) {
    const int jbase = slot << 5;
    const int nj = (WSZ - jbase) < 32 ? (WSZ - jbase) : 32;
    const float es = e4[slot];
    for (int j2 = 0; j2 < nj; ++j2) {
      const int sr = wave + jbase + j2;
      const float wj = __shfl(es, j2, 32) * inv;
      const float2 vv = *(const float2*)&vs[sr][2 * lane];
      acc.x += wj * vv.x;
      acc.y += wj * vv.y;
    }
  }

  *(float2*)(y + (size_t)pt * Cdim + h * DHd + 2 * lane) = acc;
}

// ---------------------------------------------------------------------------
extern "C" void kernel_launch(void* const* d_in, const int* in_sizes, int n_in,
                              void* d_out, int out_size, void* d_ws,
                              size_t ws_size, hipStream_t stream) {
  (void)in_sizes; (void)n_in; (void)out_size; (void)ws_size;

  const float* x     = (const float*)d_in[0];  // (T, C)
  const float* Wqkv  = (const float*)d_in[1];  // (C, 3C)
  const float* Wout  = (const float*)d_in[2];  // (C, C)
  const int*   perms = (const int*)d_in[3];    // (H, T)

  float* qkv = (float*)d_ws;  // T x 3C = 12 MB
  float* yb  = (float*)((char*)d_ws +
                        (size_t)Tdim * 3 * Cdim * sizeof(float));  // T x C

  // qkv = x @ W_qkv : M=2048, K=512, N=1536
  gemm_f32_wmma<<<dim3((3 * Cdim) / 128, Tdim / 128, 1), dim3(256, 1, 1), 0,
                  stream>>>(x, Wqkv, qkv, Cdim, 3 * Cdim);

  // permuted windowed attention: grid (T/NT, H), 512-thread blocks
  attn_perm_window<<<dim3(Tdim / NT, Hn, 1), dim3(ABLK, 1, 1), 0, stream>>>(
      qkv, perms, yb);

  // out = y @ W_out : M=2048, K=512, N=512
  gemm_f32_wmma<<<dim3(Cdim / 128, Tdim / 128, 1), dim3(256, 1, 1), 0,
                  stream>>>(yb, Wout, (float*)d_out, Cdim, Cdim);
}